// GNN_14766097564095
// MI455X (gfx1250) — compile-verified
//
#include <hip/hip_runtime.h>

typedef __attribute__((ext_vector_type(2))) float v2f;
typedef __attribute__((ext_vector_type(8))) float v8f;

#define HID 64
#define WST 68  // padded LDS stride for W (4*lane % 64 distinct -> conflict-free)

// ---------------------------------------------------------------------------
// out[r, :] = (resid ? resid[r,:] : 0) + relu( in[r,:] @ W^T )
// W is [64 out][64 in] row-major. Each wave computes a 16-row x 64-col tile
// with v_wmma_f32_16x16x4_f32 (full fp32 precision for validation).
// ---------------------------------------------------------------------------
__global__ __launch_bounds__(256) void gnn_gemm_relu(
    const float* __restrict__ in, const float* __restrict__ W,
    const float* __restrict__ resid, float* __restrict__ out, int n)
{
    __shared__ float lw[64 * WST];
    const int tid = threadIdx.x;

    // Stage W into LDS (coalesced read, padded write)
    for (int i = tid; i < 64 * 64; i += 256) {
        int r = i >> 6, c = i & 63;
        lw[r * WST + c] = W[i];
    }
    __syncthreads();

    const int wave = tid >> 5;
    const int lane = tid & 31;
    const int half = lane >> 4;   // 0: K even pair, 1: K odd pair
    const int l16  = lane & 15;
    const int row0 = blockIdx.x * 128 + wave * 16;

    // A fragments: 16 K-steps of 16x4 fp32.
    // ISA layout: lanes 0-15 M=0..15 {v0=K0, v1=K1}; lanes 16-31 {v0=K2, v1=K3}
    const int  arow  = row0 + l16;
    const bool rowok = arow < n;
    const float* abase = in + (size_t)(rowok ? arow : 0) * HID + half * 2;
    v2f a[16];
#pragma unroll
    for (int s = 0; s < 16; ++s) {
        float2 t = rowok ? *(const float2*)(abase + 4 * s) : make_float2(0.f, 0.f);
        a[s].x = t.x; a[s].y = t.y;
    }

    v8f acc[4] = {};  // four 16x16 N-tiles covering cols 0..63

#pragma unroll
    for (int s = 0; s < 16; ++s) {
#pragma unroll
        for (int j = 0; j < 4; ++j) {
            // B fragment: B[k][nc] = W[nc][k]; mirror of A layout over N.
            const int nc = j * 16 + l16;
            float2 bb = *(const float2*)&lw[nc * WST + 4 * s + half * 2];
            v2f b; b.x = bb.x; b.y = bb.y;
            acc[j] = __builtin_amdgcn_wmma_f32_16x16x4_f32(
                false, a[s], false, b, (short)0, acc[j], false, false);
        }
    }

    // D layout: VGPR v holds M = v (lanes 0-15) / v+8 (lanes 16-31), N = l16
#pragma unroll
    for (int j = 0; j < 4; ++j) {
        const int colg = j * 16 + l16;
#pragma unroll
        for (int v = 0; v < 8; ++v) {
            const int rowg = row0 + v + half * 8;
            if (rowg < n) {
                float val = fmaxf(acc[j][v], 0.f);
                if (resid) val += resid[(size_t)rowg * HID + colg];
                out[(size_t)rowg * HID + colg] = val;
            }
        }
    }
}

// ---------------------------------------------------------------------------
// Guaranteed-HW fp32 atomic add (no-return form, STOREcnt-tracked; implicit
// s_wait_idle at s_endpgm covers completion, runtime release at kernel end
// makes it visible to the next kernel on the stream).
// ---------------------------------------------------------------------------
__device__ __forceinline__ void hw_atomic_add_f32(float* p, float v) {
    asm volatile("global_atomic_add_f32 %0, %1, off"
                 :
                 : "v"(p), "v"(v)
                 : "memory");
}

// ---------------------------------------------------------------------------
// t[row[e], :] += m[col[e], :]   (four edges per wave, float2 gather per lane,
// 2x global_atomic_add_f32 per lane)
// ---------------------------------------------------------------------------
__global__ __launch_bounds__(256) void gnn_scatter(
    const float* __restrict__ m, const int* __restrict__ row,
    const int* __restrict__ col, float* __restrict__ t, int ne)
{
    const int wave = (int)((blockIdx.x * blockDim.x + threadIdx.x) >> 5);
    const int lane = threadIdx.x & 31;
    const int e0 = wave * 4;
#pragma unroll
    for (int i = 0; i < 4; ++i) {
        const int e = e0 + i;
        if (e >= ne) return;
        const int r = row[e];
        const int c = col[e];
        float2 v = *(const float2*)(m + (size_t)c * HID + 2 * lane);
        float* dst = t + (size_t)r * HID + 2 * lane;
        hw_atomic_add_f32(dst,     v.x);
        hw_atomic_add_f32(dst + 1, v.y);
    }
}

// ---------------------------------------------------------------------------
// out[r,:] = x[r,:] * rsqrt(mean(x[r,:]^2) + 1e-5) * w[:]   (one row per wave)
// ---------------------------------------------------------------------------
__global__ __launch_bounds__(256) void gnn_rmsnorm(
    const float* __restrict__ x, const float* __restrict__ w,
    float* __restrict__ out, int n)
{
    const long long gid = (long long)blockIdx.x * blockDim.x + threadIdx.x;
    const int r    = (int)(gid >> 5);
    const int lane = threadIdx.x & 31;
    if (r >= n) return;

    float2 v = *(const float2*)(x + (size_t)r * HID + 2 * lane);
    float ss = v.x * v.x + v.y * v.y;
#pragma unroll
    for (int off = 16; off >= 1; off >>= 1)
        ss += __shfl_xor(ss, off, 32);
    const float inv = rsqrtf(ss * (1.0f / 64.0f) + 1e-5f);

    float2 wv = *(const float2*)(w + 2 * lane);
    float2 o; o.x = v.x * inv * wv.x; o.y = v.y * inv * wv.y;
    *(float2*)(out + (size_t)r * HID + 2 * lane) = o;
}

// ---------------------------------------------------------------------------
extern "C" void kernel_launch(void* const* d_in, const int* in_sizes, int n_in,
                              void* d_out, int out_size, void* d_ws, size_t ws_size,
                              hipStream_t stream)
{
    const float* x       = (const float*)d_in[0];
    const float* conv_w  = (const float*)d_in[1];
    const float* conv_nw = (const float*)d_in[2];
    const float* hid_w   = (const float*)d_in[3];
    const float* hid_nw  = (const float*)d_in[4];
    const int*   row     = (const int*)d_in[5];
    const int*   col     = (const int*)d_in[6];

    const int n  = in_sizes[0] / HID;   // 100000
    const int ne = in_sizes[5];         // 800000

    float* P = (float*)d_out;                 // h ping buffer (final result)
    float* Q = (float*)d_ws;                  // relu(h @ cw^T)
    float* R = Q + (size_t)n * HID;           // residual + aggregation buffer
    const size_t hbytes = (size_t)n * HID * sizeof(float);

    const dim3 blk(256);
    const int gemm_grid = (n + 127) / 128;          // 8 waves x 16 rows
    const int rms_grid  = (n + 7) / 8;              // 8 rows per block
    const int sc_grid   = ((ne + 3) / 4 + 7) / 8;   // 4 edges/wave, 8 waves/block

    const float* hin = x;
    for (int l = 0; l < 4; ++l) {
        const float* cw = conv_w + (size_t)l * HID * HID;
        const float* hw = hid_w  + (size_t)l * HID * HID;
        const float* cn = conv_nw + l * HID;
        const float* hn = hid_nw  + l * HID;

        // m = relu(h @ cw^T)
        gnn_gemm_relu<<<gemm_grid, blk, 0, stream>>>(hin, cw, nullptr, Q, n);
        // t = h  (residual base for segment_sum)
        hipMemcpyAsync(R, hin, hbytes, hipMemcpyDeviceToDevice, stream);
        // t[row[e]] += m[col[e]]
        gnn_scatter<<<sc_grid, blk, 0, stream>>>(Q, row, col, R, ne);
        // h' = rmsnorm(t)
        gnn_rmsnorm<<<rms_grid, blk, 0, stream>>>(R, cn, R, n);
        // y = h' + relu(h' @ hw^T)
        gnn_gemm_relu<<<gemm_grid, blk, 0, stream>>>(R, hw, R, P, n);
        // h'' = rmsnorm(y)
        gnn_rmsnorm<<<rms_grid, blk, 0, stream>>>(P, hn, P, n);

        hin = P;
    }
}